// Sampler_49100066128059
// MI455X (gfx1250) — compile-verified
//
#include <hip/hip_runtime.h>
#include <math.h>
#include <stdint.h>

#define THREADS 256
#define TILE 1024          // floats per async tile (256 lanes x 16B)
#define NBUF 4             // async pipeline depth
#define CAP 2048           // candidate buffer (>= 1024 + ties)
#define KSEL 1024          // max survivors (top_k <= 1024)

typedef unsigned int u32;
typedef unsigned long long u64;

// order-preserving float->uint key (ascending key == ascending float)
__device__ __forceinline__ u32 okey(float f) {
    u32 u = __float_as_uint(f);
    return (u & 0x80000000u) ? ~u : (u ^ 0x80000000u);
}
__device__ __forceinline__ float okey_inv(u32 k) {
    u32 u = (k & 0x80000000u) ? (k ^ 0x80000000u) : ~k;
    return __uint_as_float(u);
}

// wait until <= n async-to-LDS ops outstanding (per-wave ASYNCcnt)
__device__ __forceinline__ void wait_async_le(int n) {
    switch (n) {
        case 0: asm volatile("s_wait_asynccnt 0x0" ::: "memory"); break;
        case 1: asm volatile("s_wait_asynccnt 0x1" ::: "memory"); break;
        case 2: asm volatile("s_wait_asynccnt 0x2" ::: "memory"); break;
        default: asm volatile("s_wait_asynccnt 0x3" ::: "memory"); break;
    }
}

__global__ void __launch_bounds__(THREADS) k_zero(u32* p, long n) {
    long i = (long)blockIdx.x * blockDim.x + threadIdx.x;
    if (i < n) p[i] = 0u;
}

__global__ void __launch_bounds__(THREADS)
k_scatter(const int* __restrict__ prompt, const int* __restrict__ output,
          u32* pmask, u32* omask, int PLEN, int OLEN, int MW, long np, long no) {
    long i = (long)blockIdx.x * blockDim.x + threadIdx.x;
    if (i < np) {
        int b = (int)(i / PLEN);
        int tok = prompt[i];
        atomicOr(&pmask[(long)b * MW + (tok >> 5)], 1u << (tok & 31));
    } else {
        long j = i - np;
        if (j < no) {
            int b = (int)(j / OLEN);
            int tok = output[j];
            atomicOr(&omask[(long)b * MW + (tok >> 5)], 1u << (tok & 31));
        }
    }
}

// one output-token occurrence => subtract freq once (== freq * count, deterministic)
__global__ void __launch_bounds__(THREADS)
k_freq(const int* __restrict__ output, const float* __restrict__ freq,
       float* l1, int OLEN, int V, long n) {
    long i = (long)blockIdx.x * blockDim.x + threadIdx.x;
    if (i < n) {
        int b = (int)(i / OLEN);
        int tok = output[i];
        atomicAdd(&l1[(long)b * V + tok], -freq[b]);
    }
}

// one workgroup per row; async-to-LDS streaming of logits, 4-deep pipeline
__global__ void __launch_bounds__(THREADS)
k_penalty(const float* __restrict__ logits, const u32* __restrict__ pmask,
          const u32* __restrict__ omask, const float* __restrict__ presence,
          const float* __restrict__ repetition, float* __restrict__ out_l1,
          int V, int MW) {
    int b = blockIdx.x, tid = threadIdx.x;
    __shared__ float lbuf[NBUF][TILE];
    const float* row = logits + (long)b * V;
    float* orow = out_l1 + (long)b * V;
    const u32* pm = pmask + (long)b * MW;
    const u32* om = omask + (long)b * MW;
    float pres = presence[b], rp = repetition[b];
    int nf = V / TILE;

    auto issue = [&](int t, int bufi) {
        const float* g = row + (long)t * TILE + tid * 4;
        u32 laddr = (u32)(size_t)(&lbuf[bufi][tid * 4]);   // low 32 bits of LDS flat addr = LDS offset
        u64 ga = (u64)(size_t)g;
        asm volatile("global_load_async_to_lds_b128 %0, %1, off"
                     :: "v"(laddr), "v"(ga) : "memory");
    };

    int inflight = 0;
    int pro = nf < NBUF ? nf : NBUF;
    for (int t = 0; t < pro; ++t) { issue(t, t); inflight++; }

    for (int t = 0; t < nf; ++t) {
        wait_async_le(inflight - 1);          // oldest tile arrived (in-order completion)
        int bufi = t % NBUF;
        int base = t * TILE + tid * 4;        // 4 consecutive tokens share one mask word
        u32 pw = pm[base >> 5], ow = om[base >> 5];
        float xs[4], os_[4];
#pragma unroll
        for (int c = 0; c < 4; c++) xs[c] = lbuf[bufi][tid * 4 + c];
#pragma unroll
        for (int c = 0; c < 4; c++) {
            int bit = (base & 31) + c;
            bool pb = (pw >> bit) & 1u, ob = (ow >> bit) & 1u;
            float x = xs[c];
            float rr = (pb || ob) ? rp : 1.0f;
            x = (x > 0.0f) ? x / rr : x * rr;
            if (ob) x -= pres;
            os_[c] = x;
        }
        float4 o; o.x = os_[0]; o.y = os_[1]; o.z = os_[2]; o.w = os_[3];
        *(float4*)(orow + base) = o;
        inflight--;
        int t2 = t + NBUF;
        if (t2 < nf) { issue(t2, t2 % NBUF); inflight++; }
    }
    for (int v = nf * TILE + tid; v < V; v += THREADS) {   // tail (none for V=128000)
        float x = row[v];
        u32 pw = pm[v >> 5], ow = om[v >> 5];
        bool pb = (pw >> (v & 31)) & 1u, ob = (ow >> (v & 31)) & 1u;
        float rr = (pb || ob) ? rp : 1.0f;
        x = (x > 0.0f) ? x / rr : x * rr;
        if (ob) x -= pres;
        orow[v] = x;
    }
}

// per-row: entropy->temp, radix-select 1024th largest, sort candidates, find cutoff
__global__ void __launch_bounds__(THREADS)
k_select(const float* __restrict__ l1g, const float* temperatures,
         const float* dmins, const float* dmaxs, const float* dexps,
         const float* top_ps, const int* top_ks, const float* top_as,
         const float* min_ps, float* tempw, u64* cutw, int V) {
    int b = blockIdx.x, tid = threadIdx.x;
    const float* l1 = l1g + (long)b * V;

    __shared__ u32 hist[256];
    __shared__ float redm[256], redf[256], redw[256];
    __shared__ int redi[256];
    __shared__ u64 cand[CAP];
    __shared__ float ps[KSEL], ps2[KSEL];
    __shared__ float sh_temp, sh_m2, sh_s2, sh_p0;
    __shared__ u32 sh_prefix, sh_G;
    __shared__ int sh_need, scnt, sh_cnt;

    // ---- phase 1: online logsumexp stats + top-byte histogram ----
    hist[tid] = 0u; __syncthreads();
    float m = -INFINITY, s = 0.0f, w = 0.0f;
    for (int v = tid; v < V; v += THREADS) {
        float x = l1[v];
        if (x <= m) { float e = expf(x - m); s += e; w += e * x; }
        else        { float r = expf(m - x); s = s * r + 1.0f; w = w * r + x; m = x; }
        atomicAdd(&hist[okey(x) >> 24], 1u);
    }
    redm[tid] = m; redf[tid] = s; redw[tid] = w; __syncthreads();
    for (int off = THREADS >> 1; off > 0; off >>= 1) {
        if (tid < off) {
            float ma = redm[tid], sa = redf[tid], wa = redw[tid];
            float mb = redm[tid + off], sb = redf[tid + off], wb = redw[tid + off];
            if (mb > ma) { float t; t = ma; ma = mb; mb = t; t = sa; sa = sb; sb = t; t = wa; wa = wb; wb = t; }
            float r = expf(mb - ma);
            redm[tid] = ma; redf[tid] = sa + sb * r; redw[tid] = wa + wb * r;
        }
        __syncthreads();
    }
    if (tid == 0) {
        float m1 = redm[0], S = redf[0], W = redw[0];
        float logZ = m1 + logf(S);
        float entropy = logZ - W / S;
        float norm = entropy / logf((float)V);
        float dmin = dmins[b], dmax = dmaxs[b];
        float mn = fmaxf(dmin, 0.0f);
        float dyn = mn + (dmax - mn) * powf(norm, dexps[b]);
        float T = ((dmin > 0.0f) || (dmax > 0.0f)) ? dyn : temperatures[b];
        if (T == 0.0f) T = 1.0f;
        sh_temp = T; sh_m2 = m1 / T;
        int need = (V < KSEL) ? V : KSEL;
        u32 G = 0; int sel = 0;
        for (int bin = 255; bin >= 0; --bin) {
            u32 c = hist[bin];
            if ((int)c >= need) { sel = bin; break; }
            need -= (int)c; G += c;
        }
        sh_prefix = (u32)sel << 24; sh_need = need; sh_G = G;
    }
    __syncthreads();

    // ---- phase 1.5: full softmax denominator at temperature (L2-hot rescan) ----
    float invT = 1.0f / sh_temp, m2 = sh_m2;
    float s2 = 0.0f;
    for (int v = tid; v < V; v += THREADS) s2 += expf(l1[v] * invT - m2);
    redf[tid] = s2; __syncthreads();
    for (int off = THREADS >> 1; off > 0; off >>= 1) {
        if (tid < off) redf[tid] += redf[tid + off];
        __syncthreads();
    }
    if (tid == 0) sh_s2 = redf[0];
    __syncthreads();

    // ---- radix refine: bytes 2..4 (exact 1024th-largest key) ----
    for (int shift = 16; shift >= 0; shift -= 8) {
        hist[tid] = 0u; __syncthreads();
        u32 pref = sh_prefix; int hs = shift + 8;
        for (int v = tid; v < V; v += THREADS) {
            u32 k = okey(l1[v]);
            if ((k >> hs) == (pref >> hs)) atomicAdd(&hist[(k >> shift) & 255u], 1u);
        }
        __syncthreads();
        if (tid == 0) {
            int need = sh_need; u32 G = sh_G; int sel = 0;
            for (int bin = 255; bin >= 0; --bin) {
                u32 c = hist[bin];
                if ((int)c >= need) { sel = bin; break; }
                need -= (int)c; G += c;
            }
            sh_prefix = pref | ((u32)sel << shift); sh_need = need; sh_G = G;
        }
        __syncthreads();
    }

    // ---- compact candidates (key >= Kth), pack (key, ~idx) for stable tie order ----
    if (tid == 0) scnt = 0;
    __syncthreads();
    u32 Kth = sh_prefix;
    for (int v = tid; v < V; v += THREADS) {
        u32 k = okey(l1[v]);
        if (k >= Kth) {
            int p = atomicAdd(&scnt, 1);
            if (p < CAP) cand[p] = ((u64)k << 32) | (u32)(~(u32)v);
        }
    }
    __syncthreads();
    int cnt = scnt < CAP ? scnt : CAP;
    if (tid == 0) sh_cnt = cnt;
    for (int i = tid; i < CAP; i += THREADS) if (i >= cnt) cand[i] = 0ull;
    __syncthreads();

    // ---- bitonic sort descending on u64 composite ----
    for (u32 kk = 2; kk <= CAP; kk <<= 1) {
        for (u32 j = kk >> 1; j > 0; j >>= 1) {
            for (u32 i = tid; i < CAP; i += THREADS) {
                u32 ixj = i ^ j;
                if (ixj > i) {
                    u64 a = cand[i], c = cand[ixj];
                    bool up = ((i & kk) == 0);
                    if (up ? (a < c) : (a > c)) { cand[i] = c; cand[ixj] = a; }
                }
            }
            __syncthreads();
        }
    }

    // ---- probs of top-1024, exclusive cumsum, alphabet-soup mask, n_keep ----
    float s2inv = 1.0f / sh_s2;
    cnt = sh_cnt;
    float plocal[KSEL / THREADS];
#pragma unroll
    for (int jj = 0; jj < KSEL / THREADS; ++jj) {
        int i = tid + jj * THREADS;
        float p = 0.0f;
        if (i < cnt) {
            float x = okey_inv((u32)(cand[i] >> 32));
            p = expf(x * invT - m2) * s2inv;
        }
        plocal[jj] = p; ps[i] = p;
    }
    if (tid == 0) sh_p0 = plocal[0];
    __syncthreads();

    float* src = ps; float* dst = ps2;
    for (int off = 1; off < KSEL; off <<= 1) {        // inclusive Hillis-Steele scan
        for (int i = tid; i < KSEL; i += THREADS)
            dst[i] = (i >= off) ? (src[i] + src[i - off]) : src[i];
        __syncthreads();
        float* t = src; src = dst; dst = t;
    }

    float p0 = sh_p0;
    float thr = fmaxf(p0 * min_ps[b], p0 * p0 * top_as[b]);
    float topp = top_ps[b]; int topk = top_ks[b];
    int nmin = KSEL;
#pragma unroll
    for (int jj = 0; jj < KSEL / THREADS; ++jj) {
        int i = tid + jj * THREADS;
        float p = plocal[jj];
        float cum = src[i] - p;                       // exclusive prefix
        bool mask = (p < thr) || (cum > topp) || (i >= topk);
        if (i == 0) mask = false;
        if (mask && i < nmin) nmin = i;
    }
    redi[tid] = nmin; __syncthreads();
    for (int off = THREADS >> 1; off > 0; off >>= 1) {
        if (tid < off) { int o = redi[tid + off]; if (o < redi[tid]) redi[tid] = o; }
        __syncthreads();
    }
    if (tid == 0) {
        int nk = redi[0];
        if (nk < 1) nk = 1;
        if (nk > cnt) nk = cnt;
        cutw[b] = cand[nk - 1];
        tempw[b] = sh_temp;
    }
}

// in-place: keep iff composite >= cutoff, scale survivors by 1/temp, else -inf
__global__ void __launch_bounds__(THREADS)
k_final(float* io, const float* __restrict__ tempw, const u64* __restrict__ cutw, int V) {
    int b = blockIdx.y;
    long base = (long)blockIdx.x * (THREADS * 4) + (long)threadIdx.x * 4;
    float* row = io + (long)b * V;
    float invT = 1.0f / tempw[b];
    u64 cut = cutw[b];
    if (base + 3 < V) {
        float4 x = *(float4*)(row + base);
        float xs[4] = {x.x, x.y, x.z, x.w};
        float os_[4];
#pragma unroll
        for (int c = 0; c < 4; c++) {
            u64 comp = ((u64)okey(xs[c]) << 32) | (u32)(~(u32)(base + c));
            os_[c] = (comp >= cut) ? xs[c] * invT : -INFINITY;
        }
        float4 o; o.x = os_[0]; o.y = os_[1]; o.z = os_[2]; o.w = os_[3];
        *(float4*)(row + base) = o;
    } else {
        for (long v = base; v < V && v < base + 4; ++v) {
            float xc = row[v];
            u64 comp = ((u64)okey(xc) << 32) | (u32)(~(u32)v);
            row[v] = (comp >= cut) ? xc * invT : -INFINITY;
        }
    }
}

extern "C" void kernel_launch(void* const* d_in, const int* in_sizes, int n_in,
                              void* d_out, int out_size, void* d_ws, size_t ws_size,
                              hipStream_t stream) {
    const float* logits       = (const float*)d_in[0];
    const int*   prompt       = (const int*)d_in[1];
    const int*   output       = (const int*)d_in[2];
    const float* presence     = (const float*)d_in[3];
    const float* frequency    = (const float*)d_in[4];
    const float* repetition   = (const float*)d_in[5];
    const float* temperatures = (const float*)d_in[6];
    const float* dmins        = (const float*)d_in[7];
    const float* dmaxs        = (const float*)d_in[8];
    const float* dexps        = (const float*)d_in[9];
    const float* top_ps       = (const float*)d_in[10];
    const int*   top_ks       = (const int*)d_in[11];
    const float* top_as       = (const float*)d_in[12];
    const float* min_ps       = (const float*)d_in[13];

    int B = in_sizes[3];
    int V = in_sizes[0] / B;
    int PLEN = in_sizes[1] / B;
    int OLEN = in_sizes[2] / B;
    int MW = (V + 31) / 32;

    float* l1 = (float*)d_out;                 // penalized logits live in d_out
    u32* pmask = (u32*)d_ws;
    u32* omask = pmask + (size_t)B * MW;
    float* tempw = (float*)(omask + (size_t)B * MW);
    u64* cutw = (u64*)(tempw + ((B + 1) & ~1));

    long nz = 2L * B * MW;
    k_zero<<<(int)((nz + THREADS - 1) / THREADS), THREADS, 0, stream>>>(pmask, nz);

    long np = (long)B * PLEN, no = (long)B * OLEN;
    long ns = np + no;
    k_scatter<<<(int)((ns + THREADS - 1) / THREADS), THREADS, 0, stream>>>(
        prompt, output, pmask, omask, PLEN, OLEN, MW, np, no);

    k_penalty<<<B, THREADS, 0, stream>>>(logits, pmask, omask, presence, repetition, l1, V, MW);

    k_freq<<<(int)((no + THREADS - 1) / THREADS), THREADS, 0, stream>>>(
        output, frequency, l1, OLEN, V, no);

    k_select<<<B, THREADS, 0, stream>>>(l1, temperatures, dmins, dmaxs, dexps,
                                        top_ps, top_ks, top_as, min_ps, tempw, cutw, V);

    dim3 g((V + THREADS * 4 - 1) / (THREADS * 4), B);
    k_final<<<g, THREADS, 0, stream>>>(l1, tempw, cutw, V);
}